// Encoder_90640989815157
// MI455X (gfx1250) — compile-verified
//
#include <hip/hip_runtime.h>
#include <hip/hip_bf16.h>

// Problem constants (from reference)
#define BB 4096
#define SS 512
#define HH 5
#define EE 4
#define NFF 1
#define NMM 64
#define DD 5      // NF + E
#define GG 20     // 4*H gates

typedef __attribute__((ext_vector_type(2))) float v2f;
typedef __attribute__((ext_vector_type(8))) float v8f;

// ---------------------------------------------------------------------------
// Kernel 1: xg[b,s,g] = sum_d x[b,s,d]*W_ih[g,d] + (b_ih[g]+b_hh[g])
// via V_WMMA_F32_16X16X4_F32.
//   D = A x B + C  with A = W_ih (M=gate x K=d), B = x^T (K=d x N=row)
//   => lane n (<16) accumulators hold gates 0..7 of row n  (contiguous in xg)
//      lane n+16 holds gates 8..15 of row n
//   Second M-tile (zero-padded weights) produces gates 16..19.
//   K=5 padded to 8 -> two chained WMMAs.
// One wave per 16-row tile; 8 waves per 256-thread block.
// ---------------------------------------------------------------------------
__global__ __launch_bounds__(256) void xg_wmma_kernel(
    const float* __restrict__ x_num,   // (B*S)
    const int*   __restrict__ x_cat,   // (B*S)
    const float* __restrict__ embed,   // (NM, E)
    const float* __restrict__ W_ih,    // (20, 5) row-major
    const float* __restrict__ b_ih,    // (20)
    const float* __restrict__ b_hh,    // (20)
    float* __restrict__ xg)            // (B*S, 20)
{
    __shared__ float semb[NMM * EE];   // 1 KB embedding table
    __shared__ float sbias[GG];

    const int tid = threadIdx.x;
    if (tid < NMM * EE) semb[tid] = embed[tid];
    if (tid < GG)       sbias[tid] = b_ih[tid] + b_hh[tid];
    __syncthreads();

    const int lane = tid & 31;
    const int wave = tid >> 5;
    const int n    = lane & 15;          // row-within-tile (N) and gate (M) index
    const bool hi  = lane >= 16;

    const size_t tile = (size_t)blockIdx.x * 8 + wave;
    const size_t row  = tile * 16 + (size_t)n;   // flattened (b,s) row

    // ---- B matrix (x data), per-lane K pair: (0,1)/(2,3) then (4,5)/(6,7)
    const int   cat = x_cat[row];
    const float* e  = &semb[cat * EE];
    v2f Bc1, Bc2;
    if (!hi) { Bc1[0] = x_num[row]; Bc1[1] = e[0]; Bc2[0] = e[3]; Bc2[1] = 0.f; }
    else     { Bc1[0] = e[1];       Bc1[1] = e[2]; Bc2[0] = 0.f;  Bc2[1] = 0.f; }

    // ---- A matrix (weights). Tile1: gate m = n. Tile2: gate = 16+n (n<4 valid).
    const int kb = hi ? 2 : 0;
    v2f A1c1, A1c2, A2c1, A2c2;
    A1c1[0] = W_ih[n * DD + kb];
    A1c1[1] = W_ih[n * DD + kb + 1];
    A1c2[0] = hi ? 0.f : W_ih[n * DD + 4];
    A1c2[1] = 0.f;
    if (n < 4) {
        A2c1[0] = W_ih[(16 + n) * DD + kb];
        A2c1[1] = W_ih[(16 + n) * DD + kb + 1];
        A2c2[0] = hi ? 0.f : W_ih[(16 + n) * DD + 4];
    } else {
        A2c1[0] = 0.f; A2c1[1] = 0.f; A2c2[0] = 0.f;
    }
    A2c2[1] = 0.f;

    // ---- C bias: vgpr r <-> gate r + 8*hi (tile1), 16 + r (tile2, r<4, lo lanes)
    v8f C1, C2;
#pragma unroll
    for (int r = 0; r < 8; ++r) {
        C1[r] = sbias[r + (hi ? 8 : 0)];
        C2[r] = (!hi && r < 4) ? sbias[16 + r] : 0.f;
    }

    // ---- 4 chained WMMAs (K = 0..3 then 4..7 zero-padded)
    v8f D1 = __builtin_amdgcn_wmma_f32_16x16x4_f32(false, A1c1, false, Bc1,
                                                   (short)0, C1, false, false);
    D1     = __builtin_amdgcn_wmma_f32_16x16x4_f32(false, A1c2, false, Bc2,
                                                   (short)0, D1, false, false);
    v8f D2 = __builtin_amdgcn_wmma_f32_16x16x4_f32(false, A2c1, false, Bc1,
                                                   (short)0, C2, false, false);
    D2     = __builtin_amdgcn_wmma_f32_16x16x4_f32(false, A2c2, false, Bc2,
                                                   (short)0, D2, false, false);

    // ---- Stores: xg row base is 80 B (16B-aligned); gates contiguous per lane
    float* xr = xg + row * GG;
    if (!hi) {
        *(float4*)(xr +  0) = make_float4(D1[0], D1[1], D1[2], D1[3]);
        *(float4*)(xr +  4) = make_float4(D1[4], D1[5], D1[6], D1[7]);
        *(float4*)(xr + 16) = make_float4(D2[0], D2[1], D2[2], D2[3]);
    } else {
        *(float4*)(xr +  8) = make_float4(D1[0], D1[1], D1[2], D1[3]);
        *(float4*)(xr + 12) = make_float4(D1[4], D1[5], D1[6], D1[7]);
    }
}

// ---------------------------------------------------------------------------
// Activations: exp-based, saturate correctly.
// ---------------------------------------------------------------------------
__device__ __forceinline__ float sigm_f(float x)    { return 1.f / (1.f + __expf(-x)); }
__device__ __forceinline__ float tanh_f(float x)    { return 1.f - 2.f / (1.f + __expf(2.f * x)); }

// ---------------------------------------------------------------------------
// Kernel 2: per-batch-lane recurrence. W_hh held in VGPRs (100 v_fmac chain
// per step); xg streamed with b128 loads; h written each step.
// ---------------------------------------------------------------------------
__global__ __launch_bounds__(256) void lstm_rec_kernel(
    const float* __restrict__ xg,      // (B*S, 20)
    const float* __restrict__ W_hh,    // (20, 5)
    float* __restrict__ outp)          // out(B,S,5) ++ hT(B,5) ++ cT(B,5)
{
    const int b = blockIdx.x * blockDim.x + threadIdx.x;
    if (b >= BB) return;

    float whh[GG * HH];
#pragma unroll
    for (int i = 0; i < 25; ++i) {
        float4 v = ((const float4*)W_hh)[i];
        whh[4 * i + 0] = v.x; whh[4 * i + 1] = v.y;
        whh[4 * i + 2] = v.z; whh[4 * i + 3] = v.w;
    }

    float h[HH], c[HH];
#pragma unroll
    for (int j = 0; j < HH; ++j) { h[j] = 0.f; c[j] = 0.f; }

    const float4* xp = (const float4*)(xg + (size_t)b * SS * GG);
    float* op = outp + (size_t)b * SS * HH;

    for (int t = 0; t < SS; ++t) {
        float g[GG];
        float4 v0 = xp[0], v1 = xp[1], v2 = xp[2], v3 = xp[3], v4 = xp[4];
        xp += 5;
        g[0]=v0.x; g[1]=v0.y; g[2]=v0.z; g[3]=v0.w;
        g[4]=v1.x; g[5]=v1.y; g[6]=v1.z; g[7]=v1.w;
        g[8]=v2.x; g[9]=v2.y; g[10]=v2.z; g[11]=v2.w;
        g[12]=v3.x; g[13]=v3.y; g[14]=v3.z; g[15]=v3.w;
        g[16]=v4.x; g[17]=v4.y; g[18]=v4.z; g[19]=v4.w;

#pragma unroll
        for (int gi = 0; gi < GG; ++gi) {
#pragma unroll
            for (int j = 0; j < HH; ++j)
                g[gi] = fmaf(whh[gi * HH + j], h[j], g[gi]);
        }

#pragma unroll
        for (int j = 0; j < HH; ++j) {
            const float iv = sigm_f(g[j]);
            const float fv = sigm_f(g[HH + j]);
            const float gv = tanh_f(g[2 * HH + j]);
            const float ov = sigm_f(g[3 * HH + j]);
            c[j] = fmaf(fv, c[j], iv * gv);
            h[j] = ov * tanh_f(c[j]);
            op[(size_t)t * HH + j] = h[j];
        }
    }

    float* hT = outp + (size_t)BB * SS * HH;
    float* cT = hT + (size_t)BB * HH;
#pragma unroll
    for (int j = 0; j < HH; ++j) {
        hT[(size_t)b * HH + j] = h[j];
        cT[(size_t)b * HH + j] = c[j];
    }
}

// ---------------------------------------------------------------------------
// Fallback (only if d_ws can't hold the 160 MB xg buffer): fused scalar LSTM.
// W_ih/bias/embed in LDS (uniform-address broadcast reads), W_hh in VGPRs.
// ---------------------------------------------------------------------------
__global__ __launch_bounds__(256) void lstm_fused_kernel(
    const float* __restrict__ x_num,
    const int*   __restrict__ x_cat,
    const float* __restrict__ embed,
    const float* __restrict__ W_ih,
    const float* __restrict__ W_hh,
    const float* __restrict__ b_ih,
    const float* __restrict__ b_hh,
    float* __restrict__ outp)
{
    __shared__ float swih[GG * DD];
    __shared__ float sbias[GG];
    __shared__ float semb[NMM * EE];

    const int tid = threadIdx.x;
    if (tid < GG * DD)  swih[tid]  = W_ih[tid];
    if (tid < GG)       sbias[tid] = b_ih[tid] + b_hh[tid];
    if (tid < NMM * EE) semb[tid]  = embed[tid];
    __syncthreads();

    const int b = blockIdx.x * blockDim.x + tid;
    if (b >= BB) return;

    float whh[GG * HH];
#pragma unroll
    for (int i = 0; i < 25; ++i) {
        float4 v = ((const float4*)W_hh)[i];
        whh[4 * i + 0] = v.x; whh[4 * i + 1] = v.y;
        whh[4 * i + 2] = v.z; whh[4 * i + 3] = v.w;
    }

    float h[HH], c[HH];
#pragma unroll
    for (int j = 0; j < HH; ++j) { h[j] = 0.f; c[j] = 0.f; }

    const float* xn = x_num + (size_t)b * SS;
    const int*   xc = x_cat + (size_t)b * SS;
    float* op = outp + (size_t)b * SS * HH;

    for (int t = 0; t < SS; ++t) {
        const float xv = xn[t];
        const float* e = &semb[xc[t] * EE];
        float g[GG];
#pragma unroll
        for (int gi = 0; gi < GG; ++gi) {
            float acc = sbias[gi];
            acc = fmaf(swih[gi * DD + 0], xv,  acc);
            acc = fmaf(swih[gi * DD + 1], e[0], acc);
            acc = fmaf(swih[gi * DD + 2], e[1], acc);
            acc = fmaf(swih[gi * DD + 3], e[2], acc);
            acc = fmaf(swih[gi * DD + 4], e[3], acc);
#pragma unroll
            for (int j = 0; j < HH; ++j)
                acc = fmaf(whh[gi * HH + j], h[j], acc);
            g[gi] = acc;
        }
#pragma unroll
        for (int j = 0; j < HH; ++j) {
            const float iv = sigm_f(g[j]);
            const float fv = sigm_f(g[HH + j]);
            const float gv = tanh_f(g[2 * HH + j]);
            const float ov = sigm_f(g[3 * HH + j]);
            c[j] = fmaf(fv, c[j], iv * gv);
            h[j] = ov * tanh_f(c[j]);
            op[(size_t)t * HH + j] = h[j];
        }
    }

    float* hT = outp + (size_t)BB * SS * HH;
    float* cT = hT + (size_t)BB * HH;
#pragma unroll
    for (int j = 0; j < HH; ++j) {
        hT[(size_t)b * HH + j] = h[j];
        cT[(size_t)b * HH + j] = c[j];
    }
}

extern "C" void kernel_launch(void* const* d_in, const int* in_sizes, int n_in,
                              void* d_out, int out_size, void* d_ws, size_t ws_size,
                              hipStream_t stream) {
    const float* x_num = (const float*)d_in[0];
    const int*   x_cat = (const int*)  d_in[1];
    const float* embed = (const float*)d_in[2];
    const float* W_ih  = (const float*)d_in[3];
    const float* W_hh  = (const float*)d_in[4];
    const float* b_ih  = (const float*)d_in[5];
    const float* b_hh  = (const float*)d_in[6];
    float* outp = (float*)d_out;

    const size_t xg_bytes = (size_t)BB * SS * GG * sizeof(float); // 160 MB

    if (ws_size >= xg_bytes) {
        float* xg = (float*)d_ws;
        // (B*S)/16 tiles, 8 waves (tiles) per 256-thread block
        const int tiles  = (BB * SS) / 16;       // 131072
        const int blocks = tiles / 8;            // 16384
        xg_wmma_kernel<<<blocks, 256, 0, stream>>>(x_num, x_cat, embed,
                                                   W_ih, b_ih, b_hh, xg);
        lstm_rec_kernel<<<BB / 256, 256, 0, stream>>>(xg, W_hh, outp);
    } else {
        lstm_fused_kernel<<<BB / 256, 256, 0, stream>>>(x_num, x_cat, embed,
                                                        W_ih, W_hh, b_ih, b_hh,
                                                        outp);
    }
}